// DMAQer_DVD_21706764714596
// MI455X (gfx1250) — compile-verified
//
#include <hip/hip_runtime.h>
#include <hip/hip_bf16.h>

// ---------------------------------------------------------------------------
// DMAQ/DVD mixer, fused single kernel for MI455X (gfx1250, wave32, WMMA).
//
//  * adv_q = (wf*q+v) - (wf*mq+v) = wf*(q-mq): the v hyper-net branch
//    (w1v,b1v,w2v,b2v) cancels algebraically and is skipped entirely.
//  * Big GEMMs on the fp32 matrix pipe: V_WMMA_F32_16X16X4_F32 (exact fp32).
//  * Weights are staged into LDS pre-swizzled into WMMA B-fragment order
//    ([k-step][n-tile][lane][2]) so every B operand is ONE ds_load_b64 into
//    an even-aligned VGPR pair (no v_mov marshalling), and the 12 tile loads
//    of a K-step share a single address register with immediate offsets.
//  * 8 waves/block (2 per SIMD), LDS = 323,872 B <= 320 KB/WGP.
// ---------------------------------------------------------------------------

typedef float v2f __attribute__((ext_vector_type(2)));
typedef float v4f __attribute__((ext_vector_type(4)));
typedef float v8f __attribute__((ext_vector_type(8)));

#define N_AGENTS   8
#define STATE_DIM  168
#define RNN_DIM    64
#define GAT_DIM    32
#define N_HEADS    4
#define HYPER      64
#define NCOMB      192              // HYPER + N_HEADS*GAT_DIM (fused w1f|Wd)
#define KS1        (STATE_DIM / 4)  // 42 K-steps, stage-1 GEMM
#define NT1        12               // 12 N-tiles (192/16), stage-1
#define KS2        (RNN_DIM / 4)    // 16 K-steps, GAT GEMM
#define NT2        8                // 8 N-tiles (128/16), GAT
#define ROWS_PER_WAVE   16
#define WAVES_PER_BLOCK 8
#define THREADS         (WAVES_PER_BLOCK * 32)
#define ROWS_PER_BLOCK  (ROWS_PER_WAVE * WAVES_PER_BLOCK)

// LDS layout (floats)
#define SH_W1D   (KS1 * NT1 * 32 * 2)  // 32256  frag-ordered [w1f|Wd]
#define SH_WG    (KS2 * NT2 * 32 * 2)  // 8192   frag-ordered Wg
#define SH_W2F   (HYPER * N_AGENTS)    // 512    [64][8]
#define SH_MISC  (64 + 8 + 128 + 256)  // b1f, b2f, bd, att_a[4][64]
#define SHARED_FLOATS (SH_W1D + SH_WG + SH_W2F + SH_MISC)      // 41416
// per-wave: WD[16][128]=2048 | dyn max(S 2688, H 1024 / HP 2048)=2688
//           | advq[16][8]=128 | wsc[2][4][8]=64 | tsc[2][8]=16
#define PW_FLOATS 4944
#define LDS_TOTAL_FLOATS (SHARED_FLOATS + WAVES_PER_BLOCK * PW_FLOATS) // 80968

static __device__ __forceinline__ v8f wmma4(v2f a, v2f b, v8f c) {
    // (neg_a, A, neg_b, B, c_mod, C, reuse_a, reuse_b)
    return __builtin_amdgcn_wmma_f32_16x16x4_f32(
        false, a, false, b, (short)0, c, false, false);
}

extern "C" __global__ __launch_bounds__(THREADS)
void dmaq_dvd_fused(const float* __restrict__ agent_qs,
                    const float* __restrict__ states,
                    const float* __restrict__ max_q_i,
                    const float* __restrict__ hidden,
                    const float* __restrict__ w1f, const float* __restrict__ b1f,
                    const float* __restrict__ w2f, const float* __restrict__ b2f,
                    const float* __restrict__ Wg,  const float* __restrict__ att_a,
                    const float* __restrict__ Wd,  const float* __restrict__ bd,
                    float* __restrict__ out)
{
    extern __shared__ float lds[];
    float* sW1D = lds;                        // B-fragments, stage-1
    float* sWG  = sW1D + SH_W1D;              // B-fragments, GAT
    float* sW2F = sWG + SH_WG;                // [64][8]
    float* sB1F = sW2F + SH_W2F;              // [64]
    float* sB2F = sB1F + 64;                  // [8]
    float* sBD  = sB2F + 8;                   // [128]
    float* sATT = sBD + 128;                  // [4][64]  (a_i | a_j per head)
    float* perw = sATT + 256;

    const int tid  = threadIdx.x;
    const int wave = tid >> 5;
    const int lane = tid & 31;
    const int lo   = lane & 15;               // M/N index within fragment
    const int hi   = lane >> 4;               // K-half selector

    // ---- stage weights into LDS in WMMA B-fragment order ----
    // frag element pair for (kstep s, tile t, lane L):
    //   { W[4s + 2*(L>>4)    ][16t + (L&15)],
    //     W[4s + 2*(L>>4) + 1][16t + (L&15)] }
    for (int i = tid; i < KS1 * NT1 * 32; i += THREADS) {
        const int s  = i / (NT1 * 32);
        const int r  = i - s * (NT1 * 32);
        const int t  = r >> 5;
        const int ln = r & 31;
        const int k0 = 4 * s + 2 * (ln >> 4);
        const int n  = 16 * t + (ln & 15);
        v2f e;
        if (t < 4) {                       // w1f columns (n in [0,64))
            e.x = w1f[k0 * HYPER + n];
            e.y = w1f[(k0 + 1) * HYPER + n];
        } else {                           // Wd columns (n-64 in [0,128))
            e.x = Wd[k0 * 128 + (n - HYPER)];
            e.y = Wd[(k0 + 1) * 128 + (n - HYPER)];
        }
        ((v2f*)sW1D)[i] = e;
    }
    for (int i = tid; i < KS2 * NT2 * 32; i += THREADS) {
        const int s  = i >> 8;             // / (NT2*32)
        const int r  = i & 255;
        const int t  = r >> 5;
        const int ln = r & 31;
        const int k0 = 4 * s + 2 * (ln >> 4);
        const int n  = 16 * t + (ln & 15);
        v2f e;
        e.x = Wg[k0 * 128 + n];
        e.y = Wg[(k0 + 1) * 128 + n];
        ((v2f*)sWG)[i] = e;
    }
    for (int i = tid; i < HYPER * N_AGENTS; i += THREADS) sW2F[i] = w2f[i];
    for (int i = tid; i < 256; i += THREADS) sATT[i] = att_a[i];
    if (tid < 64)  sB1F[tid] = b1f[tid];
    if (tid < 8)   sB2F[tid] = b2f[tid];
    if (tid < 128) sBD[tid]  = bd[tid];
    __syncthreads();

    float* uWD   = perw + wave * PW_FLOATS;   // [16][128], persists all chunks
    float* uDYN  = uWD + 2048;                // S[16][168] -> Y[16][64] -> H/HP
    float* uADVQ = uDYN + 2688;               // [16][8]
    float* uWSC  = uADVQ + 128;               // [2][4][8]
    float* uTSC  = uWSC + 64;                 // [2][8]

    const int R = blockIdx.x * ROWS_PER_BLOCK + wave * ROWS_PER_WAVE;

    // ---- stage S tile (16 rows x 168, contiguous in global) ----
    {
        const float* src = states + (size_t)R * STATE_DIM;
        #pragma unroll
        for (int i = 0; i < (ROWS_PER_WAVE * STATE_DIM / 4) / 32; ++i)
            ((v4f*)uDYN)[lane + 32 * i] = ((const v4f*)src)[lane + 32 * i];
    }
    __syncthreads();

    // ---- fused GEMM: S(16x168) x [w1f|Wd](168x192) -> 12 C-tiles ----
    v8f acc[NT1];
    #pragma unroll
    for (int t = 0; t < NT1; ++t)
        #pragma unroll
        for (int e = 0; e < 8; ++e) acc[t][e] = 0.0f;

    {
        const float* arow = uDYN + lo * STATE_DIM + 2 * hi;
        const v2f*   bfrg = (const v2f*)sW1D + lane;
        for (int s = 0; s < KS1; ++s) {
            v2f a = *(const v2f*)(arow + 4 * s);       // ds_load_b64
            #pragma unroll
            for (int t = 0; t < NT1; ++t)
                acc[t] = wmma4(a, bfrg[t * 32], acc[t]); // ds_load_b64 each
            bfrg += NT1 * 32;
        }
    }

    // dump Y (tiles 0..3) over dead S region; dump WD (tiles 4..11) + bd
    #pragma unroll
    for (int t = 0; t < 4; ++t)
        #pragma unroll
        for (int v = 0; v < 8; ++v)
            uDYN[(v + 8 * hi) * HYPER + t * 16 + lo] = acc[t][v];
    #pragma unroll
    for (int t = 0; t < 8; ++t)
        #pragma unroll
        for (int v = 0; v < 8; ++v)
            uWD[(v + 8 * hi) * 128 + t * 16 + lo] = acc[t + 4][v] + sBD[t * 16 + lo];
    __syncthreads();

    // ---- hyper-net epilogue: w_final and adv_q = wf*(q-mq) ----
    {
        const int row = lane >> 1;
        const int a0  = (lane & 1) * 4;
        float accq[4] = {0.f, 0.f, 0.f, 0.f};
        for (int k = 0; k < HYPER; ++k) {
            float y = uDYN[row * HYPER + k] + sB1F[k];
            y = y > 0.f ? y : 0.f;
            #pragma unroll
            for (int a = 0; a < 4; ++a) accq[a] += y * sW2F[k * 8 + a0 + a];
        }
        const size_t qb = (size_t)(R + row) * N_AGENTS;
        #pragma unroll
        for (int a = 0; a < 4; ++a) {
            float wf = fabsf(accq[a] + sB2F[a0 + a]) + 1e-10f;
            uADVQ[row * 8 + a0 + a] =
                wf * (agent_qs[qb + a0 + a] - max_q_i[qb + a0 + a]);
        }
    }
    __syncthreads();

    // ---- per 2-row chunk: hp = H(16x64) x Wg(64x128), then GAT epilogue ----
    // uH occupies uDYN[0..1024); uHP overwrites uDYN[0..2048) after the
    // K-loop (H is dead by then; same-wave LDS ops are in-order).
    float* uH  = uDYN;          // [16][64]   (row2*8+agent, d)
    float* uHP = uDYN;          // [16][128]
    for (int c = 0; c < 8; ++c) {
        const float* hsrc = hidden + (size_t)(R + 2 * c) * N_AGENTS * RNN_DIM;
        #pragma unroll
        for (int i = 0; i < 8; ++i)
            ((v4f*)uH)[lane + 32 * i] = ((const v4f*)hsrc)[lane + 32 * i];
        if (c < 7)  // prefetch next chunk's H tile (global_prefetch_b8)
            __builtin_prefetch(hsrc + 1024 + lane * 32, 0, 0);
        __syncthreads();

        v8f hacc[NT2];
        #pragma unroll
        for (int t = 0; t < NT2; ++t)
            #pragma unroll
            for (int e = 0; e < 8; ++e) hacc[t][e] = 0.0f;

        {
            const float* arow = uH + lo * RNN_DIM + 2 * hi;
            const v2f*   bfrg = (const v2f*)sWG + lane;
            for (int s = 0; s < KS2; ++s) {
                v2f a = *(const v2f*)(arow + 4 * s);
                #pragma unroll
                for (int t = 0; t < NT2; ++t)
                    hacc[t] = wmma4(a, bfrg[t * 32], hacc[t]);
                bfrg += NT2 * 32;
            }
        }
        #pragma unroll
        for (int t = 0; t < NT2; ++t)
            #pragma unroll
            for (int v = 0; v < 8; ++v)
                uHP[(v + 8 * hi) * 128 + t * 16 + lo] = hacc[t][v];
        __syncthreads();

        // GAT epilogue: lane -> (row2, head, sub); sub owns 2 query agents
        {
            const int row2 = lane >> 4;
            const int head = (lane >> 2) & 3;
            const int sub  = lane & 3;
            const float* ai  = sATT + head * 64;
            const float* aj  = ai + GAT_DIM;
            const float* hpb = uHP + row2 * 8 * 128 + head * GAT_DIM;
            const float* wdr = uWD + (2 * c + row2) * 128 + head * GAT_DIM;

            float ej[8];
            #pragma unroll
            for (int j = 0; j < 8; ++j) {
                float s = 0.f;
                for (int d = 0; d < GAT_DIM; ++d) s += hpb[j * 128 + d] * aj[d];
                ej[j] = s;
            }
            #pragma unroll
            for (int ii = 0; ii < 2; ++ii) {
                const int i = sub * 2 + ii;
                float ei = 0.f;
                for (int d = 0; d < GAT_DIM; ++d) ei += hpb[i * 128 + d] * ai[d];
                float z[8], zmax = -1e30f;
                #pragma unroll
                for (int j = 0; j < 8; ++j) {
                    float e = ei + ej[j];
                    e = e > 0.f ? e : 0.2f * e;      // leaky relu
                    z[j] = e;
                    zmax = e > zmax ? e : zmax;
                }
                float p[8], ps = 0.f;
                #pragma unroll
                for (int j = 0; j < 8; ++j) { p[j] = __expf(z[j] - zmax); ps += p[j]; }
                const float inv = 1.f / ps;
                float wdot = 0.f;
                for (int d = 0; d < GAT_DIM; ++d) {
                    float g = 0.f;
                    #pragma unroll
                    for (int j = 0; j < 8; ++j) g += p[j] * hpb[j * 128 + d];
                    g *= inv;
                    g = g > 0.f ? g : (__expf(g) - 1.f);   // elu
                    wdot += wdr[d] * g;
                }
                uWSC[(row2 * 4 + head) * 8 + i] = fabsf(wdot);
            }
        }
        __syncthreads();
        if (lane < 16) {
            const int row2 = lane >> 3, n = lane & 7;
            float m = 0.25f * (uWSC[(row2 * 4 + 0) * 8 + n] +
                               uWSC[(row2 * 4 + 1) * 8 + n] +
                               uWSC[(row2 * 4 + 2) * 8 + n] +
                               uWSC[(row2 * 4 + 3) * 8 + n]);
            uTSC[row2 * 8 + n] = uADVQ[(2 * c + row2) * 8 + n] * (m - 1.0f);
        }
        __syncthreads();
        if (lane < 2) {
            float s = 0.f;
            #pragma unroll
            for (int n = 0; n < 8; ++n) s += uTSC[lane * 8 + n];
            out[R + 2 * c + lane] = s;
        }
        __syncthreads();
    }
}

extern "C" void kernel_launch(void* const* d_in, const int* in_sizes, int n_in,
                              void* d_out, int out_size, void* d_ws, size_t ws_size,
                              hipStream_t stream) {
    const float* agent_qs = (const float*)d_in[0];
    const float* states   = (const float*)d_in[1];
    const float* max_q_i  = (const float*)d_in[2];
    const float* hidden   = (const float*)d_in[3];
    const float* w1f      = (const float*)d_in[4];
    const float* b1f      = (const float*)d_in[5];
    const float* w2f      = (const float*)d_in[6];
    const float* b2f      = (const float*)d_in[7];
    // d_in[8..11] = w1v,b1v,w2v,b2v: cancel in adv_q = wf*(q - mq); unused.
    const float* Wg       = (const float*)d_in[12];
    const float* att_a    = (const float*)d_in[13];
    const float* Wd       = (const float*)d_in[14];
    const float* bd       = (const float*)d_in[15];

    const int rows   = in_sizes[0] / N_AGENTS;          // 65536
    const int blocks = rows / ROWS_PER_BLOCK;           // 512
    const size_t shmem = LDS_TOTAL_FLOATS * sizeof(float);  // 323,872 B

    dmaq_dvd_fused<<<blocks, THREADS, shmem, stream>>>(
        agent_qs, states, max_q_i, hidden,
        w1f, b1f, w2f, b2f, Wg, att_a, Wd, bd, (float*)d_out);
}